// Attention_78726750536063
// MI455X (gfx1250) — compile-verified
//
#include <hip/hip_runtime.h>

#define BB 32
#define NN 577
#define CC 768
#define HH 12
#define HD 64
#define M_TOT (BB * NN)          // 18464, divisible by 16
#define SCALE 0.125f             // 1/sqrt(64)
#define NEG_INF (-__builtin_huge_valf())
#define NT ((NN + 15) / 16)      // 37 key tiles
#define KVPITCH 68               // 64 + 4 float pad -> conflict-free ds reads

typedef __attribute__((ext_vector_type(2))) float v2f;
typedef __attribute__((ext_vector_type(8))) float v8f;
typedef __attribute__((ext_vector_type(4))) unsigned int v4u;
typedef __attribute__((ext_vector_type(4))) int v4i;
typedef __attribute__((ext_vector_type(8))) int v8i;

#if defined(__has_builtin)
#if __has_builtin(__builtin_amdgcn_tensor_load_to_lds) && \
    __has_builtin(__builtin_amdgcn_s_wait_tensorcnt)
#define HAVE_TDM 1
#endif
#endif

__device__ __forceinline__ v8f wmma_f32(v2f a, v2f b, v8f c) {
  // D = A(16x4) * B(4x16) + C(16x16), fp32 throughout
  return __builtin_amdgcn_wmma_f32_16x16x4_f32(false, a, false, b, (short)0, c,
                                               false, false);
}

__device__ __forceinline__ v8f v8f_zero() {
  v8f z;
#pragma unroll
  for (int i = 0; i < 8; ++i) z[i] = 0.0f;
  return z;
}

#ifdef HAVE_TDM
// 2D TDM load: tile (tile0 x tile1) of a (tdim0 x tdim1) fp32 tensor with row
// stride `stride0` (elements) into LDS at byte offset lds_addr, padding the
// LDS destination by 4 DWORDs after every 64 DWORDs (row pitch 68 floats).
// D# bitfields per CDNA5 ISA 8.3/8.4.
__device__ __forceinline__ void tdm_load_2d(unsigned lds_addr,
                                            const void* gaddr, unsigned tdim0,
                                            unsigned tdim1, unsigned stride0,
                                            unsigned tile0, unsigned tile1) {
  const unsigned long long ga = (unsigned long long)(size_t)gaddr;
  v4u g0;
  g0[0] = 1u;                                    // count=1, user descriptor
  g0[1] = lds_addr;                              // lds_addr [63:32]
  g0[2] = (unsigned)(ga & 0xFFFFFFFFu);          // global_addr [95:64]
  g0[3] = (unsigned)((ga >> 32) & 0x01FFFFFFu)   // global_addr [120:96]
          | (2u << 30);                          // type=2 ("image")
  v8i g1;
  g1[0] = (int)((2u << 16)      // data_size = 4 bytes
                | (1u << 20)    // pad_enable
                | (5u << 22)    // pad_interval: 64 DWORDs
                | (3u << 25));  // pad_amount: 4 DWORDs
  g1[1] = (int)((tdim0 & 0xFFFFu) << 16);
  g1[2] = (int)(((tdim0 >> 16) & 0xFFFFu) | ((tdim1 & 0xFFFFu) << 16));
  g1[3] = (int)(((tdim1 >> 16) & 0xFFFFu) | ((tile0 & 0xFFFFu) << 16));
  g1[4] = (int)(tile1 & 0xFFFFu);                // tile_dim1 (tile_dim2 = 0)
  g1[5] = (int)stride0;                          // tensor_dim0_stride lo
  g1[6] = 0;
  g1[7] = 0;
  v4i z4;
  z4[0] = z4[1] = z4[2] = z4[3] = 0;
  v8i z8;
#pragma unroll
  for (int i = 0; i < 8; ++i) z8[i] = 0;
  // 6-arg toolchain form: (g0, g1, g2, g3, g4, cpol)
  __builtin_amdgcn_tensor_load_to_lds(g0, g1, z4, z4, z8, 0);
}
#endif

// ---------------------------------------------------------------------------
// Kernel 1: qkv = x @ Wqkv^T   (x: [M,768], Wqkv: [2304,768])
// Scatter into q/k/v buffers laid out [B,H,N,HD].
// Grid: (M/16, 2304/256), block 128 = 4 waves; each wave: 16 rows x 64 cols.
// Register double-buffered K-loop so loads overlap the WMMA chain.
// ---------------------------------------------------------------------------
__global__ __launch_bounds__(128) void qkv_kernel(
    const float* __restrict__ x, const float* __restrict__ Wqkv,
    float* __restrict__ qb, float* __restrict__ kb, float* __restrict__ vb) {
  const int lane = threadIdx.x & 31;
  const int wave = threadIdx.x >> 5;
  const int half = lane >> 4;
  const int l16 = lane & 15;
  const int rbase = blockIdx.x * 16;
  const int cbase = blockIdx.y * 256 + wave * 64;

  v8f acc[4];
#pragma unroll
  for (int cc = 0; cc < 4; ++cc) acc[cc] = v8f_zero();

  const float* arow = x + (size_t)(rbase + l16) * CC + half * 2;
  const float* brow = Wqkv + (size_t)(cbase + l16) * CC + half * 2;

  v2f a_cur = *(const v2f*)(arow);
  v2f b_cur[4];
#pragma unroll
  for (int cc = 0; cc < 4; ++cc)
    b_cur[cc] = *(const v2f*)(brow + (size_t)cc * 16 * CC);

  for (int kc = 0; kc < CC - 4; kc += 4) {
    v2f a_nxt = *(const v2f*)(arow + kc + 4);
    v2f b_nxt[4];
#pragma unroll
    for (int cc = 0; cc < 4; ++cc)
      b_nxt[cc] = *(const v2f*)(brow + (size_t)cc * 16 * CC + kc + 4);
#pragma unroll
    for (int cc = 0; cc < 4; ++cc) acc[cc] = wmma_f32(a_cur, b_cur[cc], acc[cc]);
    a_cur = a_nxt;
#pragma unroll
    for (int cc = 0; cc < 4; ++cc) b_cur[cc] = b_nxt[cc];
  }
#pragma unroll
  for (int cc = 0; cc < 4; ++cc) acc[cc] = wmma_f32(a_cur, b_cur[cc], acc[cc]);

  // cbase is a multiple of 64 -> all 64 columns of this wave share (s, h)
  const int o64 = cbase >> 6;
  const int sel = o64 / HH;  // 0=q, 1=k, 2=v
  const int h = o64 % HH;
  float* dst = (sel == 0) ? qb : (sel == 1) ? kb : vb;

#pragma unroll
  for (int i = 0; i < 8; ++i) {
    const int m = rbase + i + half * 8;
    const int b = m / NN;
    const int n = m % NN;
    float* drow = dst + (((size_t)b * HH + h) * NN + n) * HD;
#pragma unroll
    for (int cc = 0; cc < 4; ++cc) drow[cc * 16 + l16] = acc[cc][i];
  }
}

// ---------------------------------------------------------------------------
// Kernel 2: per (b,h): scores = q k^T * scale + prev (written out),
// online softmax, O = softmax(scores) @ v  -> attn buffer [B,N,C].
// K/V tiles staged in LDS once per block (TDM DMA, double-buffered) instead
// of 4x-redundant per-wave global gathers.
// Grid: (ceil(N/64), H, B), block 128 = 4 waves; wave = 16 query rows.
// ---------------------------------------------------------------------------
__global__ __launch_bounds__(128) void attn_kernel(
    const float* __restrict__ qb, const float* __restrict__ kb,
    const float* __restrict__ vb, const float* __restrict__ prev,
    float* __restrict__ scores, float* __restrict__ attn) {
  __shared__ float kls[2][16 * KVPITCH];
  __shared__ float vls[2][16 * KVPITCH];
  __shared__ float pshare[4][16][17];

  const int lane = threadIdx.x & 31;
  const int wave = threadIdx.x >> 5;
  const int half = lane >> 4;
  const int l16 = lane & 15;

  const int b = blockIdx.z;
  const int h = blockIdx.y;
  const int q0 = blockIdx.x * 64 + wave * 16;

  const size_t bh = (size_t)b * HH + h;
  const float* qp = qb + bh * (size_t)NN * HD;
  const float* kp = kb + bh * (size_t)NN * HD;
  const float* vp = vb + bh * (size_t)NN * HD;
  const float* pp = prev + bh * (size_t)NN * NN;
  float* sp = scores + bh * (size_t)NN * NN;

  // Preload Q as 16 A-fragments (covers the full hd=64 K-dimension)
  v2f qa[16];
  {
    int r = q0 + l16;
    if (r >= NN) r = NN - 1;
    const float* qr = qp + (size_t)r * HD + half * 2;
#pragma unroll
    for (int j = 0; j < 16; ++j) qa[j] = *(const v2f*)(qr + 4 * j);
  }

  v8f O[4];
#pragma unroll
  for (int dc = 0; dc < 4; ++dc) O[dc] = v8f_zero();
  float mrow[8], lrow[8];
#pragma unroll
  for (int i = 0; i < 8; ++i) {
    mrow[i] = NEG_INF;
    lrow[i] = 0.0f;
  }

#ifdef HAVE_TDM
  if (wave == 0) {
    // prologue: stage tile 0 (OOB rows zero-filled via tensor_dim1 bound)
    tdm_load_2d((unsigned)(size_t)&kls[0][0], kp, HD, NN, HD, HD, 16);
    tdm_load_2d((unsigned)(size_t)&vls[0][0], vp, HD, NN, HD, HD, 16);
  }
#endif

  for (int kt = 0; kt < NT; ++kt) {
    const int n0 = kt * 16;
#ifdef HAVE_TDM
    const int bi = kt & 1;
    if (wave == 0) {
      if (kt + 1 < NT) {
        const int n1 = (kt + 1) * 16;
        tdm_load_2d((unsigned)(size_t)&kls[bi ^ 1][0], kp + (size_t)n1 * HD,
                    HD, (unsigned)(NN - n1), HD, HD, 16);
        tdm_load_2d((unsigned)(size_t)&vls[bi ^ 1][0], vp + (size_t)n1 * HD,
                    HD, (unsigned)(NN - n1), HD, HD, 16);
        __builtin_amdgcn_s_wait_tensorcnt(2);  // tile kt landed; kt+1 in flight
      } else {
        __builtin_amdgcn_s_wait_tensorcnt(0);  // last tile landed
      }
    }
    __syncthreads();  // tile kt visible to all waves
#else
    const int bi = 0;
    __syncthreads();  // previous tile fully consumed
    for (int idx = threadIdx.x; idx < 16 * HD; idx += 128) {
      const int r = idx >> 6, c = idx & 63;
      const int gr = n0 + r;
      kls[0][r * KVPITCH + c] = (gr < NN) ? kp[(size_t)gr * HD + c] : 0.0f;
      vls[0][r * KVPITCH + c] = (gr < NN) ? vp[(size_t)gr * HD + c] : 0.0f;
    }
    __syncthreads();
#endif
    const float* kbuf = &kls[bi][0];
    const float* vbuf = &vls[bi][0];

    // ---- S = Q @ K^T (16 fp32 WMMAs over hd=64), K frags from LDS ----
    v8f s = v8f_zero();
#pragma unroll
    for (int j = 0; j < 16; ++j) {
      v2f kf = *(const v2f*)(kbuf + l16 * KVPITCH + 4 * j + half * 2);
      s = wmma_f32(qa[j], kf, s);
    }

    // ---- S = S*scale + prev; emit scores; mask OOB columns ----
    const int cg = n0 + l16;
    const int cclamp = (cg < NN) ? cg : NN - 1;
#pragma unroll
    for (int i = 0; i < 8; ++i) {
      const int r = q0 + i + half * 8;
      const int rc = (r < NN) ? r : NN - 1;
      const float pv = pp[(size_t)rc * NN + cclamp];
      const float val = s[i] * SCALE + pv;
      if (r < NN && cg < NN) sp[(size_t)r * NN + cg] = val;
      s[i] = (cg < NN) ? val : NEG_INF;
    }

    // ---- online softmax (reduce across the 16 lanes holding one row) ----
    float alpha[8];
#pragma unroll
    for (int i = 0; i < 8; ++i) {
      float mx = s[i];
      mx = fmaxf(mx, __shfl_xor(mx, 1, 16));
      mx = fmaxf(mx, __shfl_xor(mx, 2, 16));
      mx = fmaxf(mx, __shfl_xor(mx, 4, 16));
      mx = fmaxf(mx, __shfl_xor(mx, 8, 16));
      const float mnew = fmaxf(mrow[i], mx);
      const float a = __expf(mrow[i] - mnew);
      const float p = __expf(s[i] - mnew);
      float rs = p;
      rs += __shfl_xor(rs, 1, 16);
      rs += __shfl_xor(rs, 2, 16);
      rs += __shfl_xor(rs, 4, 16);
      rs += __shfl_xor(rs, 8, 16);
      lrow[i] = lrow[i] * a + rs;
      mrow[i] = mnew;
      alpha[i] = a;
      s[i] = p;
    }

    // rescale accumulator
#pragma unroll
    for (int dc = 0; dc < 4; ++dc)
#pragma unroll
      for (int i = 0; i < 8; ++i) O[dc][i] *= alpha[i];

    // ---- stage P (C-layout) -> LDS -> A-layout frags (same-wave DS RAW) ----
#pragma unroll
    for (int i = 0; i < 8; ++i) pshare[wave][i + half * 8][l16] = s[i];

    // ---- O += P @ V (16 fp32 WMMAs), V frags from LDS ----
#pragma unroll
    for (int j = 0; j < 4; ++j) {
      v2f pa;
      pa.x = pshare[wave][l16][4 * j + half * 2];
      pa.y = pshare[wave][l16][4 * j + half * 2 + 1];
      const int kk = 4 * j + half * 2;
#pragma unroll
      for (int dc = 0; dc < 4; ++dc) {
        v2f vf;
        vf.x = vbuf[kk * KVPITCH + dc * 16 + l16];
        vf.y = vbuf[(kk + 1) * KVPITCH + dc * 16 + l16];
        O[dc] = wmma_f32(pa, vf, O[dc]);
      }
    }
#ifdef HAVE_TDM
    __syncthreads();  // tile kt consumed by all waves before buffer reuse
#endif
  }

  // ---- finalize: attn[b, r, h*64 + d] = O / l ----
#pragma unroll
  for (int i = 0; i < 8; ++i) {
    const int r = q0 + i + half * 8;
    if (r >= NN) continue;
    const float inv = 1.0f / lrow[i];
    float* drow = attn + ((size_t)b * NN + r) * CC + h * HD;
#pragma unroll
    for (int dc = 0; dc < 4; ++dc) drow[dc * 16 + l16] = O[dc][i] * inv;
  }
}

// ---------------------------------------------------------------------------
// Kernel 3: out = attn @ Wproj^T + bproj   (attn: [M,768], Wproj: [768,768])
// Grid: (M/16, 768/256), block 128 = 4 waves; wave: 16 rows x 64 cols.
// ---------------------------------------------------------------------------
__global__ __launch_bounds__(128) void proj_kernel(
    const float* __restrict__ attn, const float* __restrict__ Wproj,
    const float* __restrict__ bproj, float* __restrict__ out) {
  const int lane = threadIdx.x & 31;
  const int wave = threadIdx.x >> 5;
  const int half = lane >> 4;
  const int l16 = lane & 15;
  const int rbase = blockIdx.x * 16;
  const int cbase = blockIdx.y * 256 + wave * 64;

  v8f acc[4];
#pragma unroll
  for (int cc = 0; cc < 4; ++cc) acc[cc] = v8f_zero();

  const float* arow = attn + (size_t)(rbase + l16) * CC + half * 2;
  const float* brow = Wproj + (size_t)(cbase + l16) * CC + half * 2;

  v2f a_cur = *(const v2f*)(arow);
  v2f b_cur[4];
#pragma unroll
  for (int cc = 0; cc < 4; ++cc)
    b_cur[cc] = *(const v2f*)(brow + (size_t)cc * 16 * CC);

  for (int kc = 0; kc < CC - 4; kc += 4) {
    v2f a_nxt = *(const v2f*)(arow + kc + 4);
    v2f b_nxt[4];
#pragma unroll
    for (int cc = 0; cc < 4; ++cc)
      b_nxt[cc] = *(const v2f*)(brow + (size_t)cc * 16 * CC + kc + 4);
#pragma unroll
    for (int cc = 0; cc < 4; ++cc) acc[cc] = wmma_f32(a_cur, b_cur[cc], acc[cc]);
    a_cur = a_nxt;
#pragma unroll
    for (int cc = 0; cc < 4; ++cc) b_cur[cc] = b_nxt[cc];
  }
#pragma unroll
  for (int cc = 0; cc < 4; ++cc) acc[cc] = wmma_f32(a_cur, b_cur[cc], acc[cc]);

#pragma unroll
  for (int i = 0; i < 8; ++i) {
    const int m = rbase + i + half * 8;
    float* drow = out + (size_t)m * CC;
#pragma unroll
    for (int cc = 0; cc < 4; ++cc) {
      const int o = cbase + cc * 16 + l16;
      drow[o] = acc[cc][i] + bproj[o];
    }
  }
}

// ---------------------------------------------------------------------------
extern "C" void kernel_launch(void* const* d_in, const int* in_sizes, int n_in,
                              void* d_out, int out_size, void* d_ws,
                              size_t ws_size, hipStream_t stream) {
  (void)in_sizes; (void)n_in; (void)out_size; (void)ws_size;

  const float* x     = (const float*)d_in[0];  // [B,N,C]
  const float* prev  = (const float*)d_in[1];  // [B,H,N,N]
  const float* Wqkv  = (const float*)d_in[2];  // [3C,C]
  const float* Wproj = (const float*)d_in[3];  // [C,C]
  const float* bproj = (const float*)d_in[4];  // [C]

  float* out    = (float*)d_out;               // [B,N,C]
  float* scores = out + (size_t)BB * NN * CC;  // [B,H,N,N]

  const size_t QSZ = (size_t)BB * HH * NN * HD;
  float* q    = (float*)d_ws;
  float* k    = q + QSZ;
  float* v    = k + QSZ;
  float* attn = v + QSZ;                       // [B,N,C]

  qkv_kernel<<<dim3(M_TOT / 16, (3 * CC) / 256), 128, 0, stream>>>(
      x, Wqkv, q, k, v);

  attn_kernel<<<dim3((NN + 63) / 64, HH, BB), 128, 0, stream>>>(
      q, k, v, prev, scores, attn);

  proj_kernel<<<dim3(M_TOT / 16, CC / 256), 128, 0, stream>>>(
      attn, Wproj, bproj, out);
}